// leam_82360292868113
// MI455X (gfx1250) — compile-verified
//
#include <hip/hip_runtime.h>

// Problem constants (match reference)
#define B_     4096
#define S_     200
#define E_     100
#define C_     20
#define K_     51
#define PADW_  25
#define H_     (E_/2)      // 50
#define NEG_INF (-1e13f)

#define NT_    16          // N tiles padded 13 -> 16 (compile-time uniform work split)
#define T2_    336         // gpad column count in float2 units (>=306; 336%32==16 staggers banks)
#define NTV_   13          // valid N tiles (ceil(200/16))

typedef float v2f __attribute__((ext_vector_type(2)));
typedef float v8f __attribute__((ext_vector_type(8)));

__launch_bounds__(256, 1)
__global__ void leam_kernel(const int*  __restrict__ x,       // [B,S]
                            const unsigned char* __restrict__ mask, // [B,S] bool
                            const float* __restrict__ emb,    // [V,E]
                            const float* __restrict__ labelv, // [C,E]
                            const float* __restrict__ conv_w, // [C,C,K]
                            const float* __restrict__ conv_b, // [C]
                            const float* __restrict__ w1,     // [E,H]
                            const float* __restrict__ b1,     // [H]
                            const float* __restrict__ w2,     // [H,C]
                            const float* __restrict__ b2,     // [C]
                            float* __restrict__ out)          // [B,C]
{
    // gpad2[(ci>>1)][t][ci&1] : cosine sims, ci-pair interleaved so the two K-rows of a
    // B fragment are one contiguous 8-byte LDS load. 10*336*2*4 = 26.9 KB.
    __shared__ __align__(16) float gpad2[(C_/2) * T2_ * 2];
    __shared__ __align__(16) float lv[C_ * E_];        // label vectors (8 KB)
    __shared__ float nl[C_];                           // label norms
    __shared__ unsigned mbits[256];                    // per-column channel max (float bits)
    __shared__ float red[256];                         // reduction scratch
    __shared__ float bsm[256];                         // softmax weights
    __shared__ int   sidx[256];                        // cached token indices
    __shared__ float zw[8][E_ + 4];                    // per-wave pooled partials
    __shared__ float zs[E_];
    __shared__ float hs[H_];

    const int tid  = threadIdx.x;
    const int lane = tid & 31;          // wave32
    const int wave = tid >> 5;          // 8 waves
    const int b    = blockIdx.x;

    // ---------------- stage 0: stage label vectors, zero pads ----------------
    for (int i = tid; i < C_ * E_; i += 256)          lv[i]    = labelv[i];
    for (int i = tid; i < (C_/2) * T2_ * 2; i += 256) gpad2[i] = 0.0f;
    mbits[tid] = 0u;
    __syncthreads();

    if (tid < C_) {
        float s = 0.f;
        for (int e = 0; e < E_; ++e) { float v = lv[tid * E_ + e]; s += v * v; }
        nl[tid] = sqrtf(s);
    }
    __syncthreads();

    // ---------------- stage 1: gather + cosine similarity g[c,s] ----------------
    if (tid < S_) {
        const int idx = x[b * S_ + tid];
        sidx[tid] = idx;
        const float4* row = (const float4*)(emb + idx * E_);   // 400B rows, 16B aligned
        float dot[C_];
#pragma unroll
        for (int c = 0; c < C_; ++c) dot[c] = 0.f;
        float nx = 0.f;
        for (int e4 = 0; e4 < E_ / 4; ++e4) {
            const float4 v = row[e4];
            nx += v.x * v.x + v.y * v.y + v.z * v.z + v.w * v.w;
#pragma unroll
            for (int c = 0; c < C_; ++c) {
                const float4 l = *((const float4*)(lv + c * E_) + e4);
                dot[c] += l.x * v.x + l.y * v.y + l.z * v.z + l.w * v.w;
            }
        }
        nx = sqrtf(nx);
        const int t = PADW_ + tid;
#pragma unroll
        for (int c = 0; c < C_; ++c) {
            float gh = nl[c] * nx;
            gh = (gh == 0.f) ? 1e-13f : gh;
            gpad2[((c >> 1) * T2_ + t) * 2 + (c & 1)] = dot[c] / gh;
        }
    } else {
        sidx[tid] = 0;
    }
    __syncthreads();

    // ---------------- stage 2: conv as 51 shifted GEMMs on WMMA f32 16x16x4 ----------------
    // u[co,s] = sum_k sum_ci W[co,ci,k] * g[ci, s+k-25]
    // M = 20 -> 2 tiles of 16 ; K = 20 = 5x4 ; N = 256 (16 tiles, cols >=200 discarded)
    {
        const int mtile = wave & 1;     // waves split M tiles
        const int n0    = wave >> 1;    // and N tiles: nt in {n0, n0+4, n0+8, n0+12}
        const int kb    = lane >> 4;    // K half-select per ISA fragment layout
        const int ml    = lane & 15;    // A-row (m) / B-col (n)
        // Clamp A row: rows >=20 load duplicate data -> garbage only in D rows >=20,
        // which the epilogue discards. No mask, no branch, EXEC stays all-ones.
        const int coc = min(mtile * 16 + ml, C_ - 1);
        const float* __restrict__ wrow = conv_w + coc * (C_ * K_);

        v8f acc0 = {}, acc1 = {}, acc2 = {}, acc3 = {};

        for (int k = 0; k < K_; ++k) {
#pragma unroll
            for (int kk = 0; kk < 5; ++kk) {
                const int ci0 = kk * 4 + kb * 2;     // even, <=18
                v2f a;                               // A: 16x4 of W[:, ci0..ci0+1, k]
                a.x = wrow[ ci0      * K_ + k];
                a.y = wrow[(ci0 + 1) * K_ + k];
                // B: 4x16 slab of g (shifted by k); even/odd ci pair is contiguous
                const float* gb = gpad2 + ((kk * 2 + kb) * T2_ + (k + ml)) * 2;
                v2f bf0 = *(const v2f*)(gb + (n0 +  0) * 32);
                v2f bf1 = *(const v2f*)(gb + (n0 +  4) * 32);
                v2f bf2 = *(const v2f*)(gb + (n0 +  8) * 32);
                v2f bf3 = *(const v2f*)(gb + (n0 + 12) * 32);
                acc0 = __builtin_amdgcn_wmma_f32_16x16x4_f32(false, a, false, bf0, (short)0, acc0, false, false);
                acc1 = __builtin_amdgcn_wmma_f32_16x16x4_f32(false, a, false, bf1, (short)0, acc1, false, false);
                acc2 = __builtin_amdgcn_wmma_f32_16x16x4_f32(false, a, false, bf2, (short)0, acc2, false, false);
                acc3 = __builtin_amdgcn_wmma_f32_16x16x4_f32(false, a, false, bf3, (short)0, acc3, false, false);
            }
        }

        // epilogue: +bias, ReLU, per-lane max over 8 D-rows, one LDS atomicMax per tile.
        float cb[8];
        bool  cv[8];
#pragma unroll
        for (int v = 0; v < 8; ++v) {
            const int cc = mtile * 16 + v + 8 * kb;  // D row = channel
            cv[v] = (cc < C_);
            cb[v] = conv_b[min(cc, C_ - 1)];
        }
        const v8f* accs[4] = { &acc0, &acc1, &acc2, &acc3 };
#pragma unroll
        for (int i = 0; i < 4; ++i) {
            const int nt = n0 + 4 * i;               // wave-uniform
            if (nt < NTV_) {
                const int s = nt * 16 + ml;
                float lm = 0.f;
#pragma unroll
                for (int v = 0; v < 8; ++v) {
                    float u = (*accs[i])[v] + cb[v];
                    u = fmaxf(u, 0.f);
                    lm = cv[v] ? fmaxf(lm, u) : lm;
                }
                if (s < S_) atomicMax(&mbits[s], __float_as_uint(lm));
            }
        }
    }
    __syncthreads();

    // ---------------- stage 3: masked softmax over S ----------------
    float mval = NEG_INF;
    if (tid < S_) {
        const bool mk = (mask[b * S_ + tid] != 0);
        mval = mk ? __uint_as_float(mbits[tid]) : NEG_INF;
    }
    red[tid] = mval;
    __syncthreads();
    for (int off = 128; off > 0; off >>= 1) {
        if (tid < off) red[tid] = fmaxf(red[tid], red[tid + off]);
        __syncthreads();
    }
    const float mx = red[0];
    __syncthreads();
    const float ev = (tid < S_) ? __expf(mval - mx) : 0.f;
    red[tid] = ev;
    __syncthreads();
    for (int off = 128; off > 0; off >>= 1) {
        if (tid < off) red[tid] += red[tid + off];
        __syncthreads();
    }
    const float denom = red[0];
    __syncthreads();
    bsm[tid] = ev / denom;
    __syncthreads();

    // ---------------- stage 4: attention pooling z[e] = sum_s bsm[s] * xx[s,e] ----------------
    {
        float a0 = 0.f, a1 = 0.f, a2 = 0.f, a3 = 0.f;
        for (int s = wave; s < S_; s += 8) {
            const float wt = bsm[s];
            const float* row = emb + sidx[s] * E_;   // L2-resident re-gather
            a0 += wt * row[lane];
            a1 += wt * row[lane + 32];
            a2 += wt * row[lane + 64];
            if (lane < 4) a3 += wt * row[lane + 96];
        }
        zw[wave][lane]      = a0;
        zw[wave][lane + 32] = a1;
        zw[wave][lane + 64] = a2;
        if (lane < 4) zw[wave][lane + 96] = a3;
    }
    __syncthreads();
    if (tid < E_) {
        float s = 0.f;
#pragma unroll
        for (int w = 0; w < 8; ++w) s += zw[w][tid];
        zs[tid] = s;
    }
    __syncthreads();

    // ---------------- stage 5: MLP  relu(z@w1+b1)@w2+b2 ----------------
    if (tid < H_) {
        float acc = b1[tid];
        for (int e = 0; e < E_; ++e) acc += zs[e] * w1[e * H_ + tid];
        hs[tid] = fmaxf(acc, 0.f);
    }
    __syncthreads();
    if (tid < C_) {
        float acc = b2[tid];
        for (int j = 0; j < H_; ++j) acc += hs[j] * w2[j * C_ + tid];
        out[b * C_ + tid] = acc;
    }
}

extern "C" void kernel_launch(void* const* d_in, const int* in_sizes, int n_in,
                              void* d_out, int out_size, void* d_ws, size_t ws_size,
                              hipStream_t stream) {
    const int*           x      = (const int*)d_in[0];
    /* d_in[1] = x_len (unused by reference math) */
    const unsigned char* mask   = (const unsigned char*)d_in[2];
    const float*         emb    = (const float*)d_in[3];
    const float*         labelv = (const float*)d_in[4];
    const float*         conv_w = (const float*)d_in[5];
    const float*         conv_b = (const float*)d_in[6];
    const float*         w1     = (const float*)d_in[7];
    const float*         b1     = (const float*)d_in[8];
    const float*         w2     = (const float*)d_in[9];
    const float*         b2     = (const float*)d_in[10];
    float*               out    = (float*)d_out;

    leam_kernel<<<B_, 256, 0, stream>>>(x, mask, emb, labelv, conv_w, conv_b,
                                        w1, b1, w2, b2, out);
}